// LearnedQueryAttention_30983894073876
// MI455X (gfx1250) — compile-verified
//
#include <hip/hip_runtime.h>
#include <math.h>

// ---------------------------------------------------------------------------
// LearnedQueryAttention for MI455X (gfx1250, wave32)
//
// B=8, L=4096, KD=512, ED=1024, H=8, LP=L+1=4097, DK=64, DV=128
// Memory-bound: 201 MB streamed -> ~8.6us floor @ 23.3 TB/s.
// Scores computed with V_WMMA_F32_16X16X4_F32 (fp32 WMMA, exact precision).
// ---------------------------------------------------------------------------

typedef __attribute__((ext_vector_type(2))) float v2f;
typedef __attribute__((ext_vector_type(8))) float v8f;

#define B_      8
#define L_      4096
#define LP_     4097
#define KD_     512
#define ED_     1024
#define H_      8
#define DK_     64                 // KD/H
#define DV_     128                // ED/H
#define TILES_  257                // ceil(LP/16)
#define KOFF_   (LP_ * DK_)        // 262208: per-head flat stride in batch key view
#define KBATCH_ (L_ * KD_)         // 2097152: real key elems per batch (rest = k_bias)
#define VOFF_   (LP_ * DV_)        // 524416
#define VBATCH_ (L_ * ED_)         // 4194304
#define CHUNKS_ 8
#define CROWS_  513                // ceil(LP/CHUNKS)
#define LN_EPS_ 1e-5f

// ---------------------------------------------------------------------------
// Kernel 1: scores[pair][l] = (Q[h] . K[pair,l]) * sw[h,l] + sb[h,l]
// One wave handles 16 consecutive l-rows of one (b,h) pair.
// A-matrix (16x4 f32): lanes 0-15 -> M=row, VGPR0=K0 VGPR1=K1;
//                      lanes 16-31 -> M=row, VGPR0=K2 VGPR1=K3.
// B-matrix (4x16 f32): only column n==0 (lane 0 / lane 16) carries the query.
// D column 0: lane 0 holds rows 0..7 in c[0..7], lane 16 holds rows 8..15.
// ---------------------------------------------------------------------------
__global__ __launch_bounds__(256) void lqa_scores_wmma(
    const float* __restrict__ keys, const float* __restrict__ query,
    const float* __restrict__ k_bias, const float* __restrict__ sw,
    const float* __restrict__ sb, float* __restrict__ scores)
{
    int wave = (int)((blockIdx.x * blockDim.x + threadIdx.x) >> 5);
    if (wave >= B_ * H_ * TILES_) return;   // wave-uniform guard (EXEC stays all-1)
    int lane = (int)(threadIdx.x & 31);
    int pair = wave / TILES_;
    int tile = wave - pair * TILES_;
    int b = pair >> 3, h = pair & 7;
    int r  = lane & 15;                     // matrix row this lane owns
    int kh = lane >> 4;                     // 0 -> K{0,1}, 1 -> K{2,3}
    int l  = tile * 16 + r;
    int lc = l > L_ ? L_ : l;               // clamp tail rows (not stored)
    int off_row = h * KOFF_ + lc * DK_;     // per-batch flat offset of this row
    const float* kbase = keys + (size_t)b * KBATCH_;
    const float* qh    = query + h * DK_;
    float qmask = (r == 0) ? 1.0f : 0.0f;   // only column 0 of B is the query

    v8f c = {0.f, 0.f, 0.f, 0.f, 0.f, 0.f, 0.f, 0.f};
#pragma unroll
    for (int kd = 0; kd < 16; ++kd) {
        int d0  = kd * 4 + kh * 2;          // even -> 8B-aligned pair, never
        int off = off_row + d0;             // straddles an original-row boundary
        v2f a;
        if (off < KBATCH_) {
            a = *(const v2f*)(kbase + off);
        } else {                            // appended k_bias token
            int bo = off - KBATCH_;
            a.x = k_bias[bo];
            a.y = k_bias[bo + 1];
        }
        v2f bq;
        bq.x = qmask * qh[d0];
        bq.y = qmask * qh[d0 + 1];
        c = __builtin_amdgcn_wmma_f32_16x16x4_f32(false, a, false, bq,
                                                  (short)0, c, false, false);
    }

    if (r == 0) {                           // lanes 0 and 16 hold column 0
        int mbase = tile * 16 + kh * 8;
#pragma unroll
        for (int j = 0; j < 8; ++j) {
            int lo = mbase + j;
            if (lo < LP_) {
                float s = c[j] * sw[h * LP_ + lo] + sb[h * LP_ + lo];
                scores[(size_t)pair * LP_ + lo] = s;
            }
        }
    }
}

// ---------------------------------------------------------------------------
// Kernel 2: in-place softmax over LP_ entries per (b,h). One block per pair.
// ---------------------------------------------------------------------------
__global__ __launch_bounds__(256) void lqa_softmax(float* __restrict__ sc)
{
    __shared__ float red[256];
    int tid = (int)threadIdx.x;
    float* s = sc + (size_t)blockIdx.x * LP_;

    float m = -INFINITY;
    for (int i = tid; i < LP_; i += 256) m = fmaxf(m, s[i]);
    red[tid] = m; __syncthreads();
    for (int st = 128; st > 0; st >>= 1) {
        if (tid < st) red[tid] = fmaxf(red[tid], red[tid + st]);
        __syncthreads();
    }
    float mx = red[0]; __syncthreads();

    float sum = 0.f;
    for (int i = tid; i < LP_; i += 256) {
        float e = expf(s[i] - mx);
        s[i] = e;
        sum += e;
    }
    red[tid] = sum; __syncthreads();
    for (int st = 128; st > 0; st >>= 1) {
        if (tid < st) red[tid] += red[tid + st];
        __syncthreads();
    }
    float inv = 1.0f / red[0];
    for (int i = tid; i < LP_; i += 256) s[i] *= inv;
}

// ---------------------------------------------------------------------------
// Kernel 3: partial A = sum_l attn[l] * V[l, d] over a chunk of rows.
// grid = 64 pairs * 8 chunks, block = 128 threads (one column each).
// V row reads are 128 consecutive floats -> fully coalesced 512B.
// Attention weights staged through LDS in 128-row sub-blocks.
// ---------------------------------------------------------------------------
__global__ __launch_bounds__(128) void lqa_av(
    const float* __restrict__ values, const float* __restrict__ v_bias,
    const float* __restrict__ attn, float* __restrict__ partial)
{
    __shared__ float al[128];
    int wg    = (int)blockIdx.x;
    int pair  = wg >> 3;
    int chunk = wg & 7;
    int b = pair >> 3, h = pair & 7;
    int d = (int)threadIdx.x;

    int l0 = chunk * CROWS_;
    int l1 = l0 + CROWS_; if (l1 > LP_) l1 = LP_;
    const float* att   = attn + (size_t)pair * LP_;
    const float* vbase = values + (size_t)b * VBATCH_;

    float acc = 0.f;
    for (int lb = l0; lb < l1; lb += 128) {
        int n = l1 - lb; if (n > 128) n = 128;
        if (d < n) al[d] = att[lb + d];
        __syncthreads();
        for (int i = 0; i < n; ++i) {
            int off = h * VOFF_ + (lb + i) * DV_ + d;
            float v = (off < VBATCH_) ? vbase[off] : v_bias[off - VBATCH_];
            acc += al[i] * v;
        }
        __syncthreads();
    }
    partial[((size_t)pair * CHUNKS_ + chunk) * DV_ + d] = acc;
}

// ---------------------------------------------------------------------------
// Kernel 4: reduce 8 chunk-partials per (b,h,d), then LayerNorm over ED per b.
// grid = B, block = 256 (4 embed elements per thread).
// ---------------------------------------------------------------------------
__global__ __launch_bounds__(256) void lqa_ln(
    const float* __restrict__ partial, const float* __restrict__ gamma,
    const float* __restrict__ beta, float* __restrict__ out)
{
    __shared__ float red[256];
    int b = (int)blockIdx.x;
    int tid = (int)threadIdx.x;

    float a[4];
#pragma unroll
    for (int j = 0; j < 4; ++j) {
        int e = tid + j * 256;              // 0..1023 ; e = h*128 + d
        int h = e >> 7, d = e & 127;
        const float* p = partial + ((size_t)(b * H_ + h) * CHUNKS_) * DV_ + d;
        float s = 0.f;
#pragma unroll
        for (int c2 = 0; c2 < CHUNKS_; ++c2) s += p[c2 * DV_];
        a[j] = s;
    }

    float ls = a[0] + a[1] + a[2] + a[3];
    red[tid] = ls; __syncthreads();
    for (int st = 128; st > 0; st >>= 1) {
        if (tid < st) red[tid] += red[tid + st];
        __syncthreads();
    }
    float mean = red[0] * (1.0f / (float)ED_); __syncthreads();

    float lq = 0.f;
#pragma unroll
    for (int j = 0; j < 4; ++j) { float t = a[j] - mean; lq += t * t; }
    red[tid] = lq; __syncthreads();
    for (int st = 128; st > 0; st >>= 1) {
        if (tid < st) red[tid] += red[tid + st];
        __syncthreads();
    }
    float inv = rsqrtf(red[0] * (1.0f / (float)ED_) + LN_EPS_);

#pragma unroll
    for (int j = 0; j < 4; ++j) {
        int e = tid + j * 256;
        out[(size_t)b * ED_ + e] = (a[j] - mean) * inv * gamma[e] + beta[e];
    }
}

// ---------------------------------------------------------------------------
extern "C" void kernel_launch(void* const* d_in, const int* in_sizes, int n_in,
                              void* d_out, int out_size, void* d_ws, size_t ws_size,
                              hipStream_t stream)
{
    const float* keys   = (const float*)d_in[0];   // [B,L,KD]
    const float* values = (const float*)d_in[1];   // [B,L,ED]
    const float* query  = (const float*)d_in[2];   // [H,1,DK]
    const float* k_bias = (const float*)d_in[3];   // [KD]
    const float* v_bias = (const float*)d_in[4];   // [ED]
    const float* sw     = (const float*)d_in[5];   // [H,1,LP]
    const float* sb     = (const float*)d_in[6];   // [H,1,LP]
    const float* gamma  = (const float*)d_in[7];   // [ED]
    const float* beta   = (const float*)d_in[8];   // [ED]
    float* out = (float*)d_out;                    // [B,1,ED]

    float* ws_scores  = (float*)d_ws;                          // 64*4097 floats
    float* ws_partial = ws_scores + (size_t)B_ * H_ * LP_;     // 64*8*128 floats

    // K1: 64 pairs * 257 tiles = 16448 waves, 8 waves (256 threads) per block.
    int nwaves = B_ * H_ * TILES_;
    int blocks1 = (nwaves + 7) / 8;
    lqa_scores_wmma<<<blocks1, 256, 0, stream>>>(keys, query, k_bias, sw, sb, ws_scores);

    // K2: one block per (b,h) pair.
    lqa_softmax<<<B_ * H_, 256, 0, stream>>>(ws_scores);

    // K3: 64 pairs * 8 chunks.
    lqa_av<<<B_ * H_ * CHUNKS_, 128, 0, stream>>>(values, v_bias, ws_scores, ws_partial);

    // K4: one block per batch, fused partial-reduce + LayerNorm.
    lqa_ln<<<B_, 256, 0, stream>>>(ws_partial, gamma, beta, out);
}